// GraphInteractionNetwork_58248346469036
// MI455X (gfx1250) — compile-verified
//
#include <hip/hip_runtime.h>

typedef __attribute__((ext_vector_type(2))) float v2f;
typedef __attribute__((ext_vector_type(8))) float v8f;

#define NB 8
#define NP 512
#define ND 16
#define NE 64
#define NH 128
#define NN (NB * NP) // 4096 total nodes

// ---------------------------------------------------------------------------
// Wave-level 16x16 output tile GEMM using V_WMMA_F32_16X16X4_F32.
// A: 16xK row-major (lda), B: Kx16 row-major (ldb). K multiple of 4.
// Fragment layout per CDNA5 ISA 7.12.2:
//   A 16x4: lanes 0-15 -> M=lane, VGPR j holds K = 2*halfwave + j
//   B 4x16: lanes hold N=lane&15, VGPR j holds K = 2*halfwave + j
//   C/D 16x16: VGPR v holds M = v + 8*halfwave, N = lane&15
// ---------------------------------------------------------------------------
__device__ __forceinline__ v8f wmma16x16(const float* __restrict__ A, int lda,
                                         const float* __restrict__ B, int ldb,
                                         int K, v8f acc) {
  const int lane = threadIdx.x & 31;
  const int mn = lane & 15;        // row of A / col of B owned by this lane
  const int kb = (lane >> 4) * 2;  // half-wave K sub-offset
  for (int k0 = 0; k0 < K; k0 += 4) {
    v2f a, b;
    a.x = A[mn * lda + k0 + kb];
    a.y = A[mn * lda + k0 + kb + 1];
    b.x = B[(k0 + kb) * ldb + mn];
    b.y = B[(k0 + kb + 1) * ldb + mn];
    acc = __builtin_amdgcn_wmma_f32_16x16x4_f32(false, a, false, b, (short)0,
                                                acc, false, false);
  }
  return acc;
}

// ---------------------------------------------------------------------------
// Kernel 1: per-node projections through the first edge layer.
//   proj_r[n] = nodes[n] @ We1[1:17]  + be1   (receiver term, bias folded in)
//   proj_s[n] = nodes[n] @ We1[17:33]         (sender term)
// grid = (NN/16, 2), block = 256 (8 waves = 8 column tiles of H=128)
// ---------------------------------------------------------------------------
__global__ __launch_bounds__(256) void proj_kernel(
    const float* __restrict__ nodes, const float* __restrict__ We1,
    const float* __restrict__ be1, float* __restrict__ proj_r,
    float* __restrict__ proj_s) {
  const int tile = blockIdx.x;        // 16-node row tile
  const int which = blockIdx.y;       // 0 = receiver proj, 1 = sender proj
  const int wave = threadIdx.x >> 5;  // column tile (8 x 16 = 128 cols)
  const int lane = threadIdx.x & 31;

  const float* A = nodes + (size_t)tile * 16 * ND;
  const float* B = We1 + (size_t)(1 + 16 * which) * NH + 16 * wave;

  v8f acc = {};
  acc = wmma16x16(A, ND, B, NH, ND, acc);

  const int half = lane >> 4;
  const int n = lane & 15;
  const int col = 16 * wave + n;
  const float bias = (which == 0) ? be1[col] : 0.0f;
  float* dst = (which == 0) ? proj_r : proj_s;
#pragma unroll
  for (int v = 0; v < 8; ++v) {
    const int m = v + 8 * half;
    dst[((size_t)tile * 16 + m) * NH + col] = acc[v] + bias;
  }
}

// ---------------------------------------------------------------------------
// Kernel 2 (hot loop): ReLU-aggregate hidden edge activations per receiver.
//   agg_h[b,r][k] = sum_{s != r} relu(dist(s,r)*w0[k] + proj_r'[r][k] + proj_s[s][k])
// One block of 256 threads per (b, r). Distances staged in LDS once; the
// accumulation splits 512 senders across two thread halves (128 channels each).
// proj_s for a batch is 256 KB -> L2-resident, reused by 512 receiver blocks.
// ---------------------------------------------------------------------------
__global__ __launch_bounds__(256) void edge_agg_kernel(
    const float* __restrict__ nodes, const float* __restrict__ proj_r,
    const float* __restrict__ proj_s, const float* __restrict__ We1,
    float* __restrict__ agg_h) {
  __shared__ float nrsh[ND];
  __shared__ float distsh[NP];
  __shared__ float partial[256];

  const int bn = blockIdx.x;  // global receiver node index
  const int b = bn >> 9;
  const int r = bn & (NP - 1);
  const int t = threadIdx.x;

  if (t < ND) nrsh[t] = nodes[(size_t)bn * ND + t];
  __syncthreads();

#pragma unroll
  for (int i = 0; i < 2; ++i) {
    const int s = t + i * 256;
    const float* ns = nodes + ((size_t)b * NP + s) * ND;
    float d2 = 0.0f;
#pragma unroll
    for (int d = 0; d < ND; ++d) {
      const float df = ns[d] - nrsh[d];
      d2 = fmaf(df, df, d2);
    }
    distsh[s] = sqrtf(d2);
  }
  __syncthreads();

  const int k = t & (NH - 1);   // hidden channel
  const int hh = t >> 7;        // sender half: [0,256) or [256,512)
  const float w0k = We1[k];     // We1 row 0 = distance weights
  const float prb = proj_r[(size_t)bn * NH + k];  // includes be1
  const float* psb = proj_s + ((size_t)b * NP) * NH;

  float acc = 0.0f;
#pragma unroll 4
  for (int i = 0; i < 256; ++i) {
    const int s = hh * 256 + i;
    const float hpre = fmaf(distsh[s], w0k, prb + psb[(size_t)s * NH + k]);
    const float rel = fmaxf(hpre, 0.0f);
    acc += (s == r) ? 0.0f : rel;  // exclude self-edge
  }
  partial[t] = acc;
  __syncthreads();
  if (t < NH) agg_h[(size_t)bn * NH + t] = partial[t] + partial[t + NH];
}

// ---------------------------------------------------------------------------
// Kernel 3: agg = agg_h @ We2 + 511*be2   (NN x 128) @ (128 x 64)
// grid = NN/32, block = 256: each block covers 2 row tiles x 4 col tiles.
// ---------------------------------------------------------------------------
__global__ __launch_bounds__(256) void agg_gemm_kernel(
    const float* __restrict__ agg_h, const float* __restrict__ We2,
    const float* __restrict__ be2, float* __restrict__ agg) {
  const int wave = threadIdx.x >> 5;
  const int lane = threadIdx.x & 31;
  const int rowtile = blockIdx.x * 2 + (wave >> 2);
  const int coltile = wave & 3;

  const float* A = agg_h + (size_t)rowtile * 16 * NH;
  const float* B = We2 + 16 * coltile;

  v8f acc = {};
  acc = wmma16x16(A, NH, B, NE, NH, acc);

  const int half = lane >> 4;
  const int n = lane & 15;
  const int col = 16 * coltile + n;
  const float bias = 511.0f * be2[col];  // P-1 edges per receiver
#pragma unroll
  for (int v = 0; v < 8; ++v) {
    const int m = v + 8 * half;
    agg[((size_t)rowtile * 16 + m) * NE + col] = acc[v] + bias;
  }
}

// ---------------------------------------------------------------------------
// Kernel 4: fused node MLP.
//   hid = relu([agg | nodes] @ Wn1 + bn1)  (16 x 128 tile, split K = 64 + 16)
//   out = hid @ Wn2 + bn2                  (16 x 16, K=128 split over 8 waves)
// grid = NN/16, block = 256 (8 waves).
// ---------------------------------------------------------------------------
__global__ __launch_bounds__(256) void node_mlp_kernel(
    const float* __restrict__ agg, const float* __restrict__ nodes,
    const float* __restrict__ Wn1, const float* __restrict__ bn1,
    const float* __restrict__ Wn2, const float* __restrict__ bn2,
    float* __restrict__ out) {
  __shared__ float hid[16 * NH];
  __shared__ float part[8 * 256];

  const int tile = blockIdx.x;
  const int wave = threadIdx.x >> 5;
  const int lane = threadIdx.x & 31;
  const int half = lane >> 4;
  const int n = lane & 15;

  // hidden layer: acc = agg_tile(16x64)@Wn1[0:64] + nodes_tile(16x16)@Wn1[64:80]
  v8f acc = {};
  acc = wmma16x16(agg + (size_t)tile * 16 * NE, NE, Wn1 + 16 * wave, NH, NE, acc);
  acc = wmma16x16(nodes + (size_t)tile * 16 * ND, ND,
                  Wn1 + (size_t)NE * NH + 16 * wave, NH, ND, acc);

  const int col = 16 * wave + n;
  const float b1 = bn1[col];
#pragma unroll
  for (int v = 0; v < 8; ++v) {
    const int m = v + 8 * half;
    hid[m * NH + col] = fmaxf(acc[v] + b1, 0.0f);
  }
  __syncthreads();

  // output layer: each wave handles a K-slice of 16, partials reduced in LDS
  v8f acc2 = {};
  acc2 = wmma16x16(hid + 16 * wave, NH, Wn2 + (size_t)(16 * wave) * ND, ND, 16,
                   acc2);
#pragma unroll
  for (int v = 0; v < 8; ++v) part[wave * 256 + v * 32 + lane] = acc2[v];
  __syncthreads();

  const int t = threadIdx.x;
  const int mo = t >> 4;  // output row 0..15
  const int no = t & 15;  // output col 0..15
  const int v = mo & 7;
  const int ls = ((mo >> 3) << 4) + no;
  float s = 0.0f;
#pragma unroll
  for (int w = 0; w < 8; ++w) s += part[w * 256 + v * 32 + ls];
  out[((size_t)tile * 16 + mo) * ND + no] = s + bn2[no];
}

// ---------------------------------------------------------------------------
extern "C" void kernel_launch(void* const* d_in, const int* in_sizes, int n_in,
                              void* d_out, int out_size, void* d_ws,
                              size_t ws_size, hipStream_t stream) {
  (void)in_sizes; (void)n_in; (void)out_size; (void)ws_size;
  const float* h = (const float*)d_in[0];    // (B, P*D) == nodes (NN x 16)
  const float* We1 = (const float*)d_in[1];  // 33 x 128
  const float* be1 = (const float*)d_in[2];  // 128
  const float* We2 = (const float*)d_in[3];  // 128 x 64
  const float* be2 = (const float*)d_in[4];  // 64
  const float* Wn1 = (const float*)d_in[5];  // 80 x 128
  const float* bn1 = (const float*)d_in[6];  // 128
  const float* Wn2 = (const float*)d_in[7];  // 128 x 16
  const float* bn2 = (const float*)d_in[8];  // 16
  float* out = (float*)d_out;

  float* ws = (float*)d_ws;
  float* proj_r = ws;                          // NN*128 (be1 folded in)
  float* proj_s = ws + (size_t)NN * NH;        // NN*128
  float* agg_h = ws + (size_t)2 * NN * NH;     // NN*128
  float* agg = ws + (size_t)3 * NN * NH;       // NN*64

  proj_kernel<<<dim3(NN / 16, 2), 256, 0, stream>>>(h, We1, be1, proj_r, proj_s);
  edge_agg_kernel<<<NN, 256, 0, stream>>>(h, proj_r, proj_s, We1, agg_h);
  agg_gemm_kernel<<<NN / 32, 256, 0, stream>>>(agg_h, We2, be2, agg);
  node_mlp_kernel<<<NN / 16, 256, 0, stream>>>(agg, h, Wn1, bn1, Wn2, bn2, out);
}